// PtModule_76166950027789
// MI455X (gfx1250) — compile-verified
//
#include <hip/hip_runtime.h>
#include <hip/hip_bf16.h>

// Fused elementwise: out = ((x + 2) * 3)^2 = (3x + 6)^2
// HBM-bound (0.375 flop/byte). Target: pure b128 NT streaming at 23.3 TB/s.
//
// Fast path: exact-cover kernel, no loop, no bounds checks, 32-bit indexing
// -> 4x global_load_b128 NT, 8 VALU (v_fma + v_mul per vec4), 4x
//    global_store_b128 NT per thread.
// Fallback: generic grid-stride kernel for shapes that don't divide evenly.

typedef float v4f __attribute__((ext_vector_type(4)));

#define BLOCK 256
#define UNROLL 4
#define CHUNK (BLOCK * UNROLL)  // vec4s per block

// ---------------- Fast path: shape divides exactly ----------------
__global__ __launch_bounds__(BLOCK) void fused_elemwise_exact(
    const float* __restrict__ in, float* __restrict__ out)
{
    const v4f* __restrict__ vin  = reinterpret_cast<const v4f*>(in);
    v4f* __restrict__       vout = reinterpret_cast<v4f*>(out);

    // 32-bit indexing: max index 16,777,215; max byte offset 256 MB - both
    // fit in u32 -> SADDR + 32-bit VOFFSET addressing, minimal VALU.
    const unsigned base = blockIdx.x * (unsigned)CHUNK + threadIdx.x;

    v4f r[UNROLL];
    #pragma unroll
    for (int k = 0; k < UNROLL; ++k) {
        r[k] = __builtin_nontemporal_load(vin + (base + (unsigned)k * BLOCK));
    }
    #pragma unroll
    for (int k = 0; k < UNROLL; ++k) {
        v4f t = r[k] * 3.0f + 6.0f;   // (x+2)*3 == 3x+6 -> v_fma
        __builtin_nontemporal_store(t * t, vout + (base + (unsigned)k * BLOCK));
    }
}

// ---------------- Generic fallback: any size ----------------
__global__ __launch_bounds__(BLOCK) void fused_elemwise_generic(
    const float* __restrict__ in, float* __restrict__ out,
    long long nvec, long long n)
{
    const v4f* __restrict__ vin  = reinterpret_cast<const v4f*>(in);
    v4f* __restrict__       vout = reinterpret_cast<v4f*>(out);

    const long long tid    = (long long)blockIdx.x * BLOCK + threadIdx.x;
    const long long stride = (long long)gridDim.x * BLOCK;

    for (long long i = tid; i < nvec; i += stride) {
        v4f x = __builtin_nontemporal_load(vin + i);
        v4f t = x * 3.0f + 6.0f;
        __builtin_nontemporal_store(t * t, vout + i);
    }

    // Scalar tail (n % 4 != 0)
    const long long tail_start = nvec * 4;
    const long long tail_count = n - tail_start;
    if (tid < tail_count) {
        float x = in[tail_start + tid];
        float t = fmaf(x, 3.0f, 6.0f);
        out[tail_start + tid] = t * t;
    }
}

extern "C" void kernel_launch(void* const* d_in, const int* in_sizes, int n_in,
                              void* d_out, int out_size, void* d_ws, size_t ws_size,
                              hipStream_t stream) {
    (void)n_in; (void)d_ws; (void)ws_size; (void)out_size;

    const float* in = reinterpret_cast<const float*>(d_in[0]);
    float* out      = reinterpret_cast<float*>(d_out);

    const long long n    = (long long)in_sizes[0];  // 67,108,864 for reference
    const long long nvec = n / 4;                   // 16,777,216

    const bool exact = (n % 4 == 0) && (nvec % CHUNK == 0) &&
                       (nvec / CHUNK <= 0x7FFFFFFFLL);

    if (exact) {
        const unsigned blocks = (unsigned)(nvec / CHUNK);  // 16,384
        fused_elemwise_exact<<<dim3(blocks), dim3(BLOCK), 0, stream>>>(in, out);
    } else {
        long long blocks = (nvec + BLOCK - 1) / BLOCK;
        if (blocks < 1) blocks = 1;
        if (blocks > 262144) blocks = 262144;  // grid-stride covers the rest
        fused_elemwise_generic<<<dim3((unsigned)blocks), dim3(BLOCK), 0, stream>>>(
            in, out, nvec, n);
    }
}